// UniqueRoleLoss_52415780880539
// MI455X (gfx1250) — compile-verified
//
#include <hip/hip_runtime.h>
#include <hip/hip_bf16.h>

#define B_    16
#define S_    256
#define NL_   66
#define VMAX_ 8
#define C_    6

typedef __attribute__((ext_vector_type(2))) float v2f;
typedef __attribute__((ext_vector_type(8))) float v8f;

struct MinPair { float m1; float m2; int i1; };

__device__ __forceinline__ MinPair combine(MinPair a, MinPair b) {
    MinPair r;
    if (b.m1 < a.m1) { r.m1 = b.m1; r.i1 = b.i1; r.m2 = fminf(a.m1, b.m2); }
    else             { r.m1 = a.m1; r.i1 = a.i1; r.m2 = fminf(b.m1, a.m2); }
    return r;
}

__global__ void url_zero(float* acc) { acc[0] = 0.0f; }

__global__ __launch_bounds__(256)
void url_main(const float* __restrict__ log_pa,
              const int*  __restrict__ v_label,
              const int*  __restrict__ v_l,
              const int*  __restrict__ orig_l,
              float*      __restrict__ acc) {
    const int b = blockIdx.x / VMAX_;
    const int v = blockIdx.x % VMAX_;
    if (v >= v_l[b]) return;                 // uniform per block: contributes 0

    const int lbl = v_label[b * VMAX_ + v];
    const int ol  = orig_l[b];
    const int j   = threadIdx.x;             // token index, S_ == blockDim.x
    const int lane = j & 31, wave = j >> 5;

    // gather the 6 core-label log-probs for (b, predicate row lbl, token j)
    const float* row = log_pa + (((size_t)b * S_ + lbl) * S_ + j) * NL_ + 1;
    float lc[C_], ln[C_];
#pragma unroll
    for (int c = 0; c < C_; ++c) lc[c] = row[c];
#pragma unroll
    for (int c = 0; c < C_; ++c)
        ln[c] = logf(fmaxf(1.0f - expf(lc[c]), 1e-6f));

    __shared__ float sm1[C_][8], sm2[C_][8];
    __shared__ int   si1[C_][8];
    __shared__ float fm1[C_], fm2[C_];
    __shared__ int   fi1[C_];
    __shared__ float wsum[16];

    // (min1, min2, argmin1) over valid tokens, per core label
#pragma unroll
    for (int c = 0; c < C_; ++c) {
        MinPair p;
        p.m1 = (j < ol) ? ln[c] : 1e30f;     // padded tokens never win
        p.m2 = 3.0e38f;
        p.i1 = j;
#pragma unroll
        for (int m = 1; m < 32; m <<= 1) {
            MinPair q;
            q.m1 = __shfl_xor(p.m1, m, 32);
            q.m2 = __shfl_xor(p.m2, m, 32);
            q.i1 = __shfl_xor(p.i1, m, 32);
            p = combine(p, q);
        }
        if (lane == 0) { sm1[c][wave] = p.m1; sm2[c][wave] = p.m2; si1[c][wave] = p.i1; }
    }
    __syncthreads();
    if (j == 0) {
#pragma unroll
        for (int c = 0; c < C_; ++c) {
            MinPair p{sm1[c][0], sm2[c][0], si1[c][0]};
            for (int w = 1; w < 8; ++w) {
                MinPair q{sm1[c][w], sm2[c][w], si1[c][w]};
                p = combine(p, q);
            }
            fm1[c] = p.m1; fm2[c] = p.m2; fi1[c] = p.i1;
        }
    }
    __syncthreads();

    // relu(log_core - min_excluding_self), valid tokens only
    float partial = 0.0f;
    if (j < ol) {
#pragma unroll
        for (int c = 0; c < C_; ++c) {
            const float mn = (j == fi1[c]) ? fm2[c] : fm1[c];
            partial += fmaxf(lc[c] - mn, 0.0f);
        }
    }

    // Wave sum via V_WMMA_F32_16X16X4_F32: A(16x4) holds the 32 partials
    // (lanes 0-15 -> K0, lanes 16-31 -> K2; K1/K3 zero), B = ones(4x16).
    // Column 0 of D: D[m,0] = p[m] + p[m+16]; lanes 0 and 16 hold M=0..7 and
    // M=8..15 of column 0 in their 8 C/D VGPRs. EXEC is all-ones here.
    v2f a;  a[0] = partial; a[1] = 0.0f;
    v2f bo; bo[0] = 1.0f;   bo[1] = 1.0f;
    v8f cz = {};
    v8f d = __builtin_amdgcn_wmma_f32_16x16x4_f32(
        /*neg_a=*/false, a, /*neg_b=*/false, bo,
        /*c_mod=*/(short)0, cz, /*reuse_a=*/false, /*reuse_b=*/false);
    float loc = 0.0f;
#pragma unroll
    for (int i = 0; i < 8; ++i) loc += d[i];

    if ((lane & 15) == 0) wsum[wave * 2 + (lane >> 4)] = loc;
    __syncthreads();
    if (j == 0) {
        float s = 0.0f;
#pragma unroll
        for (int i = 0; i < 16; ++i) s += wsum[i];
        atomicAdd(acc, s);
    }
}

__global__ void url_fin(const float* __restrict__ acc,
                        const int* __restrict__ v_l,
                        float* __restrict__ out) {
    int n = 0;
    for (int i = 0; i < B_; ++i) n += v_l[i];
    out[0] = acc[0] / (float)n;
}

extern "C" void kernel_launch(void* const* d_in, const int* in_sizes, int n_in,
                              void* d_out, int out_size, void* d_ws, size_t ws_size,
                              hipStream_t stream) {
    const float* log_pa = (const float*)d_in[0];
    const int*   v_lab  = (const int*)d_in[1];
    const int*   v_l    = (const int*)d_in[2];
    const int*   orig_l = (const int*)d_in[3];
    float* acc = (float*)d_ws;

    url_zero<<<1, 1, 0, stream>>>(acc);
    url_main<<<B_ * VMAX_, 256, 0, stream>>>(log_pa, v_lab, v_l, orig_l, acc);
    url_fin<<<1, 1, 0, stream>>>(acc, v_l, (float*)d_out);
}